// LabelSmoothingCEWithSuperLoss_87643102642248
// MI455X (gfx1250) — compile-verified
//
#include <hip/hip_runtime.h>
#include <stdint.h>

#define C_CLASSES 330
#define EPS_F 0.1f
#define LAM_F 0.25f
#define ROWS_PER_WAVE 8
#define WAVES_PER_BLOCK 8
#define PAIRS 165                    // 330 floats = 165 float2 (rows are 8B aligned: 1320 % 8 == 0)
#define SLOT_BYTES 1328              // 1320 padded to 16B multiple
#define NBUF 3                       // triple-buffered async pipeline
#define LDS_BYTES (WAVES_PER_BLOCK * NBUF * SLOT_BYTES)

// Issue one row (1320B) as 6 per-lane async b64 copies global -> LDS.
// The IOFFSET immediate is added to BOTH the global and the LDS address,
// so all issues share one base register pair. Each instruction bumps this
// wave's ASYNCcnt by 1 => exactly 6 outstanding per row batch.
__device__ __forceinline__ void async_row_to_lds(const float* __restrict__ gsrc,
                                                 uint32_t lds_off, int lane) {
  uint32_t la = lds_off + (uint32_t)(lane * 8);
  uint64_t ga = (uint64_t)(uintptr_t)gsrc + (uint64_t)(lane * 8);
  asm volatile("global_load_async_to_lds_b64 %0, %1, off th:TH_LOAD_NT"
               :: "v"(la), "v"(ga) : "memory");
  asm volatile("global_load_async_to_lds_b64 %0, %1, off offset:256 th:TH_LOAD_NT"
               :: "v"(la), "v"(ga) : "memory");
  asm volatile("global_load_async_to_lds_b64 %0, %1, off offset:512 th:TH_LOAD_NT"
               :: "v"(la), "v"(ga) : "memory");
  asm volatile("global_load_async_to_lds_b64 %0, %1, off offset:768 th:TH_LOAD_NT"
               :: "v"(la), "v"(ga) : "memory");
  asm volatile("global_load_async_to_lds_b64 %0, %1, off offset:1024 th:TH_LOAD_NT"
               :: "v"(la), "v"(ga) : "memory");
  if (lane < 5)  // pairs 160..164
    asm volatile("global_load_async_to_lds_b64 %0, %1, off offset:1280 th:TH_LOAD_NT"
                 :: "v"(la), "v"(ga) : "memory");
}

__global__ void __launch_bounds__(256)
superloss_main(const float* __restrict__ logits, const int* __restrict__ tgt,
               int B, int total_waves, double* __restrict__ acc) {
  __shared__ __align__(16) unsigned char smem_raw[LDS_BYTES];

  const int lane = (int)(threadIdx.x & 31u);
  const int wave = (int)(threadIdx.x >> 5u);
  const int gw   = (int)blockIdx.x * WAVES_PER_BLOCK + wave;

  const float tau = 5.799092654460526f;          // log(330)
  // Low 32 bits of a flat LDS pointer are the LDS byte offset (aperture in high bits).
  const uint32_t base_off =
      (uint32_t)(uintptr_t)(smem_raw) + (uint32_t)(wave * NBUF * SLOT_BYTES);
  uint32_t slot_off[NBUF];
  const float* fslot[NBUF];
#pragma unroll
  for (int k = 0; k < NBUF; ++k) {
    slot_off[k] = base_off + (uint32_t)(k * SLOT_BYTES);
    fslot[k] = (const float*)(smem_raw + wave * NBUF * SLOT_BYTES + k * SLOT_BYTES);
  }

  double smooth_acc = 0.0;
  float  my_l = tau;   // lanes with no assigned row -> y=0 -> w=0 -> super term == 0

  if (gw < total_waves) {
    int issued = 0;    // number of 6-op row batches issued so far (wave-uniform)

    // Prologue: fill up to two buffers ahead.
#pragma unroll
    for (int k = 0; k < 2; ++k) {
      long long rk = (long long)gw + (long long)total_waves * k;
      if (k < ROWS_PER_WAVE && rk < B) {
        async_row_to_lds(logits + (size_t)rk * C_CLASSES, slot_off[k], lane);
        ++issued;
      }
    }

#pragma unroll 1
    for (int j = 0; j < ROWS_PER_WAVE; ++j) {
      long long r = (long long)gw + (long long)total_waves * j;
      if (r >= B) break;
      const int buf = j % NBUF;

      // Keep the pipeline two rows deep.
      long long r2 = (long long)gw + (long long)total_waves * (j + 2);
      if ((j + 2) < ROWS_PER_WAVE && r2 < B) {
        async_row_to_lds(logits + (size_t)r2 * C_CLASSES, slot_off[(j + 2) % NBUF], lane);
        ++issued;
      }

      // Retire batch j: allow 6 ops per batch still genuinely ahead of us.
      const int ahead = issued - j - 1;          // 0, 1 or 2 (wave-uniform)
      if (ahead >= 2)      asm volatile("s_wait_asynccnt 0xc" ::: "memory");
      else if (ahead == 1) asm volatile("s_wait_asynccnt 0x6" ::: "memory");
      else                 asm volatile("s_wait_asynccnt 0x0" ::: "memory");

      const float*  rowf = fslot[buf];
      const float2* rowp = (const float2*)rowf;

      float2 v[6];
      float m = -3.4e38f, s = 0.f;
#pragma unroll
      for (int i = 0; i < 6; ++i) {
        int p = i * 32 + lane;
        if (p < PAIRS) {
          float2 t2 = rowp[p];
          v[i] = t2;
          m  = fmaxf(m, fmaxf(t2.x, t2.y));
          s += t2.x + t2.y;
        } else {
          v[i] = make_float2(0.f, 0.f);
        }
      }
#pragma unroll
      for (int off = 16; off > 0; off >>= 1) m = fmaxf(m, __shfl_xor(m, off));

      float e = 0.f;
#pragma unroll
      for (int i = 0; i < 6; ++i) {
        int p = i * 32 + lane;
        if (p < PAIRS) e += __expf(v[i].x - m) + __expf(v[i].y - m);
      }
#pragma unroll
      for (int off = 16; off > 0; off >>= 1) {
        e += __shfl_xor(e, off);
        s += __shfl_xor(s, off);
      }

      float lse = m + __logf(e);
      int   t   = tgt[r];
      float xt  = rowf[t];                        // single broadcast LDS read
      float li  = lse - xt;

      smooth_acc += (double)((float)C_CLASSES * lse - s);  // -sum_c log p_c
      if (lane == j) my_l = li;                            // park row j's loss on lane j
    }
  }

  // ---- Lane-parallel Lambert W (8 rows solved at once across lanes 0..7) ----
  float y  = 0.5f * fmaxf(-0.7357588823428847f, (my_l - tau) * (1.0f / LAM_F));
  float pp = sqrtf(fmaxf(2.0f * (2.718281828459045f * y + 1.0f), 0.0f));
  float w  = -1.0f + pp - pp * pp * (1.0f / 3.0f) + (11.0f / 72.0f) * pp * pp * pp;
  if (y >= 0.3f) w = log1pf(y);
#pragma unroll
  for (int it = 0; it < 12; ++it) {
    float ew   = __expf(w);
    float f    = w * ew - y;
    float wp1  = w + 1.0f;
    float swp1 = (fabsf(wp1) < 1e-12f) ? 1e-12f : wp1;
    float den  = ew * wp1 - (w + 2.0f) * f / (2.0f * swp1);
    float sden = (fabsf(den) < 1e-30f) ? 1e-30f : den;
    float step = (fabsf(f) < 1e-30f) ? 0.0f : f / sden;
    w -= step;
  }
  float sigma  = __expf(-w);
  float superv = (my_l - tau) * sigma + LAM_F * w * w;
#pragma unroll
  for (int off = 16; off > 0; off >>= 1) superv += __shfl_xor(superv, off);

  if (lane == 0) {
    atomicAdd(&acc[0], smooth_acc);         // global_atomic_add_f64
    atomicAdd(&acc[1], (double)superv);
  }
}

__global__ void superloss_zero(double* acc) {
  acc[0] = 0.0;
  acc[1] = 0.0;
}

__global__ void superloss_finalize(const double* __restrict__ acc,
                                   float* __restrict__ out, int B) {
  double smooth_mean = acc[0] / (double)B;
  double super_mean  = acc[1] / (double)B;
  out[0] = (float)(smooth_mean * ((double)EPS_F / (double)C_CLASSES) +
                   (1.0 - (double)EPS_F) * super_mean);
}

extern "C" void kernel_launch(void* const* d_in, const int* in_sizes, int n_in,
                              void* d_out, int out_size, void* d_ws, size_t ws_size,
                              hipStream_t stream) {
  (void)n_in; (void)out_size; (void)ws_size;
  const float* logits = (const float*)d_in[0];
  const int*   tgt    = (const int*)d_in[1];
  int B = in_sizes[1];                      // target has B elements
  double* acc = (double*)d_ws;

  int total_waves = (B + ROWS_PER_WAVE - 1) / ROWS_PER_WAVE;
  int blocks      = (total_waves + WAVES_PER_BLOCK - 1) / WAVES_PER_BLOCK;

  superloss_zero<<<1, 1, 0, stream>>>(acc);
  superloss_main<<<blocks, WAVES_PER_BLOCK * 32, 0, stream>>>(logits, tgt, B,
                                                              total_waves, acc);
  superloss_finalize<<<1, 1, 0, stream>>>(acc, (float*)d_out, B);
}